// LogisticRegressionTorchModule_72481868087920
// MI455X (gfx1250) — compile-verified
//
#include <hip/hip_runtime.h>

typedef __attribute__((ext_vector_type(2))) float v2f;
typedef __attribute__((ext_vector_type(4))) float v4f;
typedef __attribute__((ext_vector_type(8))) float v8f;

#define DD    64       // n_in
#define NS    128      // N_samples
#define LOG2E 1.44269504088896340736f

// Broadcast from a compile-time-constant lane -> v_readlane_b32 (no LDS).
__device__ __forceinline__ float bcastf(float v, int srcLane) {
    return __builtin_bit_cast(
        float, __builtin_amdgcn_readlane(__builtin_bit_cast(int, v), srcLane));
}

// ---------------------------------------------------------------------------
// Pre-kernel: build the (constant) WMMA B-tile once, in per-lane layout:
//   Bws[lane*32 + kk*2 + comp],  lane in [0,32), kk in [0,16), comp in {0,1}
// Value for lane (half = lane>>4, mn = lane&15), k0 = kk*4 + half*2 + comp:
//   mn==0 -> w_mu[k0], mn==1 -> exp(w_log_var[k0]), else 0
// Total 4 KB; runs once per launch, negligible.
// ---------------------------------------------------------------------------
__global__ void build_b_tile(const float* __restrict__ w_mu,
                             const float* __restrict__ w_lv,
                             float* __restrict__ Bws)
{
    const int t    = threadIdx.x;        // 0..1023, t == lane*32 + kk*2 + comp
    const int lane = t >> 5;
    const int kk   = (t >> 1) & 15;
    const int comp = t & 1;
    const int half = lane >> 4;
    const int mn   = lane & 15;
    const int k0   = kk * 4 + half * 2 + comp;
    float v = 0.0f;
    if (mn == 0)      v = w_mu[k0];
    else if (mn == 1) v = __builtin_amdgcn_exp2f(w_lv[k0] * LOG2E);
    Bws[t] = v;
}

// sigmoid(s*z + m) = rcp(1 + exp2(sl*z + ml)), sl=-log2e*s, ml=-log2e*m
// -> per element: v_fma + v_exp_f32 + v_add + v_rcp_f32 (TRANS co-executes)
__device__ __forceinline__ void emit_row(const v8f& c, int r, const v4f& zv,
                                         float* __restrict__ rowp, int lane)
{
    const int   src = (r >> 3) << 4;
    const float m   = bcastf(c[r & 7], src + 0);   // N=0 lane of row M=r
    const float va  = bcastf(c[r & 7], src + 1);   // N=1 lane of row M=r
    const float s   = __builtin_amdgcn_sqrtf(va);
    const float sl  = -LOG2E * s;
    const float ml  = -LOG2E * m;
    v4f o;
    o.x = __builtin_amdgcn_rcpf(1.0f + __builtin_amdgcn_exp2f(fmaf(sl, zv.x, ml)));
    o.y = __builtin_amdgcn_rcpf(1.0f + __builtin_amdgcn_exp2f(fmaf(sl, zv.y, ml)));
    o.z = __builtin_amdgcn_rcpf(1.0f + __builtin_amdgcn_exp2f(fmaf(sl, zv.z, ml)));
    o.w = __builtin_amdgcn_rcpf(1.0f + __builtin_amdgcn_exp2f(fmaf(sl, zv.w, ml)));
    // streaming 256MB output: non-temporal B128 store, keep L2 clean
    __builtin_nontemporal_store(o, (v4f*)(rowp + lane * 4));
}

// One wave (32 lanes) processes a 16-row tile of X.
// mean/var for the 16 rows come from 16 chained V_WMMA_F32_16X16X4_F32:
//   C[16x16] += A[16x4] * B[4x16],  B col0 = w_mu, col1 = exp(w_log_var).
template <bool USE_WS>
__global__ __launch_bounds__(256)
void logreg_sample_wmma(const float* __restrict__ X,
                        const float* __restrict__ w_mu,
                        const float* __restrict__ w_lv,
                        const float* __restrict__ z,
                        const float* __restrict__ Bws,
                        float* __restrict__ out,
                        int n)
{
    const int lane = threadIdx.x & 31;
    const int wave = threadIdx.x >> 5;
    const int half = lane >> 4;   // K-half selector (ISA f32 A/B layout)
    const int mn   = lane & 15;   // M for A-tile, N for B-tile

    const long row_base = ((long)blockIdx.x * 8 + wave) * 16;

    // ---- A tile: X[row_base .. row_base+15][0..63] in WMMA f32 layout ----
    // lane (half,mn): V0 = X[mn][4*kk + 2*half], V1 = X[mn][4*kk + 2*half + 1]
    long arow = row_base + mn;
    if (arow >= n) arow = n - 1;          // clamp: tail rows are never stored
    const float* Xr = X + arow * DD;

    v2f a[16];
#pragma unroll
    for (int kk = 0; kk < 16; ++kk) {
        a[kk] = *(const v2f*)(Xr + kk * 4 + half * 2);
    }

    // ---- B tile ----
    v2f b[16];
    if constexpr (USE_WS) {
        // precomputed, per-lane contiguous -> 8 x B128 loads, L0/L2 resident
        const float* bp = Bws + lane * 32;
#pragma unroll
        for (int q = 0; q < 8; ++q) {
            const v4f t = *(const v4f*)(bp + q * 4);
            b[2 * q + 0].x = t.x; b[2 * q + 0].y = t.y;
            b[2 * q + 1].x = t.z; b[2 * q + 1].y = t.w;
        }
    } else {
        // fallback: build inline (branchless loads + selects)
#pragma unroll
        for (int kk = 0; kk < 16; ++kk) {
            const int k0 = kk * 4 + half * 2;
            const v2f wm = *(const v2f*)(w_mu + k0);
            const v2f lv = *(const v2f*)(w_lv + k0);
            const float e0 = __builtin_amdgcn_exp2f(lv.x * LOG2E);
            const float e1 = __builtin_amdgcn_exp2f(lv.y * LOG2E);
            b[kk].x = (mn == 0) ? wm.x : ((mn == 1) ? e0 : 0.0f);
            b[kk].y = (mn == 0) ? wm.y : ((mn == 1) ? e1 : 0.0f);
        }
    }

    // ---- K-reduction on the matrix unit: 16 chained WMMAs (K = 64) ----
    v8f c = {0.f, 0.f, 0.f, 0.f, 0.f, 0.f, 0.f, 0.f};
#pragma unroll
    for (int kk = 0; kk < 16; ++kk) {
        c = __builtin_amdgcn_wmma_f32_16x16x4_f32(
                /*neg_a=*/false, a[kk], /*neg_b=*/false, b[kk],
                /*c_mod=*/(short)0, c, /*reuse_a=*/false, /*reuse_b=*/false);
    }
    // C element (M = v + 8*(lane/16), N = lane%16):
    //   N==0 -> mean of row M, N==1 -> var of row M.

    // Private z chunk: lane covers samples [4*lane, 4*lane+3]
    const v4f zv = *(const v4f*)(z + lane * 4);

    // ---- 16 coalesced row stores: 32 lanes x float4 = 512B per row ----
    float* outp = out + row_base * NS;
    if (row_base + 16 <= (long)n) {
        // full tile: no per-row guard in the hot path
#pragma unroll
        for (int r = 0; r < 16; ++r) {
            emit_row(c, r, zv, outp + (long)r * NS, lane);
        }
    } else {
        // tail tile: wave-uniform per-row guard
#pragma unroll
        for (int r = 0; r < 16; ++r) {
            if (row_base + r < n) {
                emit_row(c, r, zv, outp + (long)r * NS, lane);
            }
        }
    }
}

extern "C" void kernel_launch(void* const* d_in, const int* in_sizes, int n_in,
                              void* d_out, int out_size, void* d_ws, size_t ws_size,
                              hipStream_t stream)
{
    const float* X    = (const float*)d_in[0];
    const float* w_mu = (const float*)d_in[1];
    const float* w_lv = (const float*)d_in[2];
    const float* z    = (const float*)d_in[3];
    float* out        = (float*)d_out;

    const int n = in_sizes[0] / DD;              // 500000 rows
    const int rows_per_block = 8 * 16;           // 8 waves x 16-row tiles
    const int blocks = (n + rows_per_block - 1) / rows_per_block;

    if (ws_size >= 4096) {
        float* Bws = (float*)d_ws;
        build_b_tile<<<1, 1024, 0, stream>>>(w_mu, w_lv, Bws);
        logreg_sample_wmma<true><<<blocks, 256, 0, stream>>>(
            X, w_mu, w_lv, z, Bws, out, n);
    } else {
        logreg_sample_wmma<false><<<blocks, 256, 0, stream>>>(
            X, w_mu, w_lv, z, nullptr, out, n);
    }
}